// Model_flow_72834055406080
// MI455X (gfx1250) — compile-verified
//
#include <hip/hip_runtime.h>

// Problem constants (from reference): B=16, H=448, W=1024, C=3
#define B_DIM 16
#define H_DIM 448
#define W_DIM 1024
#define C_DIM 3

static constexpr int kHW   = H_DIM * W_DIM;   // 458752
static constexpr int kNPix = B_DIM * kHW;     // 7340032

// Tile decomposition: each block privatizes a 64x16 pixel tile (+4px halo) in LDS.
static constexpr int TW = 64;                 // tile width (pixels)
static constexpr int TH = 16;                 // tile height (pixels)
static constexpr int R  = 4;                  // halo radius (N(0,1) flow => ~6e-5 spill rate)
static constexpr int LW = TW + 2 * R;         // 72 cells per row
static constexpr int LH = TH + 2 * R;         // 24 rows
// Row stride padded to an ODD dword count (72*3=216 -> 219) so the two rows a
// wave covers hit disjoint LDS bank sets (mod 4) within each ds_add_f32.
static constexpr int LSTRIDE    = LW * C_DIM + 3;     // 219 dwords per row
static constexpr int LDS_FLOATS = LSTRIDE * LH;       // 5256 floats (21.0 KB), /4 = 1314

static constexpr int TILES_X = W_DIM / TW;    // 16
static constexpr int TILES_Y = H_DIM / TH;    // 28
static constexpr int NBLOCKS = B_DIM * TILES_X * TILES_Y;  // 7168
static constexpr int NTHREADS = 256;          // 8 waves (wave32); 4 px/thread -> 1024 px/tile

// ---------------------------------------------------------------------------
// Kernel 1: zero-fill output (harness poisons d_out with 0xAA). float4 stores
// also pre-warm the 88MB output footprint into the 192MB L2 so the splat
// kernel's global atomics are L2-resident RMWs.
// ---------------------------------------------------------------------------
__global__ __launch_bounds__(256) void zero_out_kernel(float4* __restrict__ out, int n4) {
    int i = blockIdx.x * 256 + threadIdx.x;
    if (i < n4) {
        out[i] = make_float4(0.f, 0.f, 0.f, 0.f);
    }
}

// ---------------------------------------------------------------------------
// Kernel 2: LDS-privatized bilinear forward splat.
//   x = ix + flo_x ; y = iy + flo_y  (exact reduction of the reference math)
// Fast path (one branch per pixel): whole 2x2 footprint inside LDS tile ->
// 12 unconditional ds_add_f32 from one base address + immediate offsets.
// Slow path (rare): guarded global atomics per corner.
// Then flush tile+halo to global with fp32 atomics, skipping all-zero cells
// and out-of-image halo cells.
// ---------------------------------------------------------------------------
__global__ __launch_bounds__(NTHREADS) void splat_kernel(const float* __restrict__ U,
                                                         const float* __restrict__ flo,
                                                         float* __restrict__ out) {
    __shared__ float tile[LDS_FLOATS];

    const int tid = threadIdx.x;
    const int blk = blockIdx.x;

    // Block -> (batch, tile_y, tile_x)
    const int tx = blk % TILES_X;
    const int ty = (blk / TILES_X) % TILES_Y;
    const int b  = blk / (TILES_X * TILES_Y);

    const int tile_x0 = tx * TW;
    const int tile_y0 = ty * TH;
    const int lds_x0  = tile_x0 - R;   // global x of LDS cell (.,0)
    const int lds_y0  = tile_y0 - R;   // global y of LDS cell (0,.)

    // ---- Phase 1: zero LDS tile (vectorized b128 stores; 1314 = 256*5 + 34) ----
    {
        float4* t4 = reinterpret_cast<float4*>(tile);
        const float4 z = make_float4(0.f, 0.f, 0.f, 0.f);
        for (int i = tid; i < LDS_FLOATS / 4; i += NTHREADS) {
            t4[i] = z;
        }
    }
    __syncthreads();

    // ---- Phase 2: splat 4 consecutive pixels per thread ----
    const int row  = tid >> 4;            // 0..15
    const int col4 = (tid & 15) * 4;      // 0..60
    const int iy   = tile_y0 + row;
    const int ix0  = tile_x0 + col4;
    const int p0   = (b * H_DIM + iy) * W_DIM + ix0;   // multiple of 4

    // Prefetch the tile directly below (emits global_prefetch_b8).
    {
        const int pf = p0 + TH * W_DIM;
        if (pf < kNPix) {
            __builtin_prefetch(U   + (size_t)pf * C_DIM, 0, 1);
            __builtin_prefetch(flo + (size_t)pf * 2,     0, 1);
        }
    }

    // Coalesced b128 loads: 12 floats of U, 8 floats of flo (16B-aligned).
    const float4* U4 = reinterpret_cast<const float4*>(U + (size_t)p0 * C_DIM);
    const float4  u0 = U4[0], u1 = U4[1], u2 = U4[2];
    const float4* F4 = reinterpret_cast<const float4*>(flo + (size_t)p0 * 2);
    const float4  f0 = F4[0], f1 = F4[1];

    const float uc[4][C_DIM] = {
        {u0.x, u0.y, u0.z},
        {u0.w, u1.x, u1.y},
        {u1.z, u1.w, u2.x},
        {u2.y, u2.z, u2.w},
    };
    const float fxv[4] = {f0.x, f0.z, f1.x, f1.z};
    const float fyv[4] = {f0.y, f0.w, f1.y, f1.w};

    float* const out_b = out + (size_t)b * kHW * C_DIM;

#pragma unroll
    for (int k = 0; k < 4; ++k) {
        const float x = (float)(ix0 + k) + fxv[k];
        const float y = (float)iy        + fyv[k];

        const float x0f = floorf(x);
        const float y0f = floorf(y);
        const int   x0  = (int)x0f;
        const int   y0  = (int)y0f;
        const int   x1  = x0 + 1;
        const int   y1  = y0 + 1;

        const float ax = x - x0f;   // frac x
        const float ay = y - y0f;   // frac y
        const float bx = 1.0f - ax;
        const float by = 1.0f - ay;

        const bool x0i = (unsigned)x0 < (unsigned)W_DIM;
        const bool x1i = (unsigned)x1 < (unsigned)W_DIM;
        const bool y0i = (unsigned)y0 < (unsigned)H_DIM;
        const bool y1i = (unsigned)y1 < (unsigned)H_DIM;

        const float wa = (x0i && y0i) ? bx * by : 0.0f;  // (y0, x0)
        const float wb = (x0i && y1i) ? bx * ay : 0.0f;  // (y1, x0)
        const float wc = (x1i && y0i) ? ax * by : 0.0f;  // (y0, x1)
        const float wd = (x1i && y1i) ? ax * ay : 0.0f;  // (y1, x1)

        const float c0 = uc[k][0];
        const float c1 = uc[k][1];
        const float c2 = uc[k][2];

        const int lx0 = x0 - lds_x0;
        const int ly0 = y0 - lds_y0;

        // One branch per pixel: entire 2x2 footprint inside the LDS region?
        if ((unsigned)lx0 < (unsigned)(LW - 1) && (unsigned)ly0 < (unsigned)(LH - 1)) {
            // Hot path: 12 branch-free ds_add_f32; corner deltas become
            // immediate DS offsets off one base address. Zero weights (OOB
            // corners at the image border) harmlessly add 0 to halo cells.
            float* p = tile + ly0 * LSTRIDE + lx0 * C_DIM;
            __hip_atomic_fetch_add(p + 0,               c0 * wa, __ATOMIC_RELAXED, __HIP_MEMORY_SCOPE_WORKGROUP);
            __hip_atomic_fetch_add(p + 1,               c1 * wa, __ATOMIC_RELAXED, __HIP_MEMORY_SCOPE_WORKGROUP);
            __hip_atomic_fetch_add(p + 2,               c2 * wa, __ATOMIC_RELAXED, __HIP_MEMORY_SCOPE_WORKGROUP);
            __hip_atomic_fetch_add(p + C_DIM + 0,       c0 * wc, __ATOMIC_RELAXED, __HIP_MEMORY_SCOPE_WORKGROUP);
            __hip_atomic_fetch_add(p + C_DIM + 1,       c1 * wc, __ATOMIC_RELAXED, __HIP_MEMORY_SCOPE_WORKGROUP);
            __hip_atomic_fetch_add(p + C_DIM + 2,       c2 * wc, __ATOMIC_RELAXED, __HIP_MEMORY_SCOPE_WORKGROUP);
            __hip_atomic_fetch_add(p + LSTRIDE + 0,     c0 * wb, __ATOMIC_RELAXED, __HIP_MEMORY_SCOPE_WORKGROUP);
            __hip_atomic_fetch_add(p + LSTRIDE + 1,     c1 * wb, __ATOMIC_RELAXED, __HIP_MEMORY_SCOPE_WORKGROUP);
            __hip_atomic_fetch_add(p + LSTRIDE + 2,     c2 * wb, __ATOMIC_RELAXED, __HIP_MEMORY_SCOPE_WORKGROUP);
            __hip_atomic_fetch_add(p + LSTRIDE + C_DIM + 0, c0 * wd, __ATOMIC_RELAXED, __HIP_MEMORY_SCOPE_WORKGROUP);
            __hip_atomic_fetch_add(p + LSTRIDE + C_DIM + 1, c1 * wd, __ATOMIC_RELAXED, __HIP_MEMORY_SCOPE_WORKGROUP);
            __hip_atomic_fetch_add(p + LSTRIDE + C_DIM + 2, c2 * wd, __ATOMIC_RELAXED, __HIP_MEMORY_SCOPE_WORKGROUP);
        } else {
            // Rare spill (|flow| > ~4 or footprint clips the halo edge):
            // guarded global atomics. w != 0 guarantees in-image coords.
            auto gcorner = [&](int yy, int xx, float w) {
                if (w != 0.0f) {
                    float* p = out_b + ((size_t)yy * W_DIM + xx) * C_DIM;
                    __hip_atomic_fetch_add(p + 0, c0 * w, __ATOMIC_RELAXED, __HIP_MEMORY_SCOPE_AGENT);
                    __hip_atomic_fetch_add(p + 1, c1 * w, __ATOMIC_RELAXED, __HIP_MEMORY_SCOPE_AGENT);
                    __hip_atomic_fetch_add(p + 2, c2 * w, __ATOMIC_RELAXED, __HIP_MEMORY_SCOPE_AGENT);
                }
            };
            gcorner(y0, x0, wa);
            gcorner(y1, x0, wb);
            gcorner(y0, x1, wc);
            gcorner(y1, x1, wd);
        }
    }

    __syncthreads();

    // ---- Phase 3: flush tile+halo to global memory ----
    for (int i = tid; i < LW * LH; i += NTHREADS) {
        const int ly = i / LW;
        const int lx = i - ly * LW;
        const int gx = lds_x0 + lx;
        const int gy = lds_y0 + ly;

        const float* c = tile + ly * LSTRIDE + lx * C_DIM;
        const float v0 = c[0];
        const float v1 = c[1];
        const float v2 = c[2];

        const bool inb = ((unsigned)gx < (unsigned)W_DIM) && ((unsigned)gy < (unsigned)H_DIM);
        if (inb && (v0 != 0.0f || v1 != 0.0f || v2 != 0.0f)) {
            float* p = out_b + ((size_t)gy * W_DIM + gx) * C_DIM;
            __hip_atomic_fetch_add(p + 0, v0, __ATOMIC_RELAXED, __HIP_MEMORY_SCOPE_AGENT);
            __hip_atomic_fetch_add(p + 1, v1, __ATOMIC_RELAXED, __HIP_MEMORY_SCOPE_AGENT);
            __hip_atomic_fetch_add(p + 2, v2, __ATOMIC_RELAXED, __HIP_MEMORY_SCOPE_AGENT);
        }
    }
}

// ---------------------------------------------------------------------------
// Host-side launcher. Inputs: d_in[0] = U (float32), d_in[1] = flo (float32).
// Output: d_out = [B,H,W,C] float32.
// ---------------------------------------------------------------------------
extern "C" void kernel_launch(void* const* d_in, const int* in_sizes, int n_in,
                              void* d_out, int out_size, void* d_ws, size_t ws_size,
                              hipStream_t stream) {
    (void)in_sizes; (void)n_in; (void)d_ws; (void)ws_size;

    const float* U   = (const float*)d_in[0];
    const float* flo = (const float*)d_in[1];
    float*       out = (float*)d_out;

    // Zero-init output (out_size = 22,020,096 floats, divisible by 4).
    const int n4 = out_size / 4;
    const int zblocks = (n4 + 255) / 256;
    zero_out_kernel<<<zblocks, 256, 0, stream>>>((float4*)out, n4);

    // Privatized splat: one block per 64x16 tile, 4 pixels per thread.
    splat_kernel<<<NBLOCKS, NTHREADS, 0, stream>>>(U, flo, out);
}